// GraphConv1TPK_79250736546091
// MI455X (gfx1250) — compile-verified
//
#include <hip/hip_runtime.h>
#include <math.h>

typedef float v2f __attribute__((ext_vector_type(2)));
typedef float v8f __attribute__((ext_vector_type(8)));

#define NTN   131072      // total nodes
#define NE    1048576     // edges
#define BG    128         // graphs
#define NN    1024        // nodes per graph
#define FF    128         // features
#define KK    820         // kept per graph
#define BKN   (BG*KK)     // 104960 kept nodes total
#define NCLS  10

// ---------------------------------------------------------------- zero
__global__ void k_zero(float* p, long n) {
    long i = (long)blockIdx.x * blockDim.x + threadIdx.x;
    long stride = (long)gridDim.x * blockDim.x;
    float4 z = {0.f, 0.f, 0.f, 0.f};
    for (long j = i; j * 4 < n; j += stride) ((float4*)p)[j] = z;
}

// ------------------------------------------------------ edge scatter-add
// one wave per edge; lane handles 4 contiguous features (float4 gather + 4 atomics)
__global__ void k_scatter(const float* __restrict__ H, const int* __restrict__ src,
                          const int* __restrict__ dst, float* Agg, int nE) {
    int lane = threadIdx.x & 31;
    int wid  = (blockIdx.x * blockDim.x + threadIdx.x) >> 5;
    int nw   = (gridDim.x * blockDim.x) >> 5;
    for (int e = wid; e < nE; e += nw) {
        int s = src[e], d = dst[e];
        float4 v = ((const float4*)(H + (long)s * FF))[lane];
        float* a = Agg + (long)d * FF + lane * 4;
        atomicAdd(a + 0, v.x); atomicAdd(a + 1, v.y);
        atomicAdd(a + 2, v.z); atomicAdd(a + 3, v.w);
    }
}

__global__ void k_scatter_masked(const float* __restrict__ HP, const int* __restrict__ src,
                                 const int* __restrict__ dst, const int* __restrict__ nidx,
                                 float* Agg, int nE) {
    int lane = threadIdx.x & 31;
    int wid  = (blockIdx.x * blockDim.x + threadIdx.x) >> 5;
    int nw   = (gridDim.x * blockDim.x) >> 5;
    for (int e = wid; e < nE; e += nw) {
        int ns = nidx[src[e]], nd = nidx[dst[e]];
        if (ns >= 0 && nd >= 0) {
            float4 v = ((const float4*)(HP + (long)ns * FF))[lane];
            float* a = Agg + (long)nd * FF + lane * 4;
            atomicAdd(a + 0, v.x); atomicAdd(a + 1, v.y);
            atomicAdd(a + 2, v.z); atomicAdd(a + 3, v.w);
        }
    }
}

// -------------------------------------------- fused GraphConv GEMM (WMMA fp32)
// Out = relu(A1 @ W1 + A2 @ W2 + bias)   with A1/A2: [M,128], W: [128,128]
// Out may alias A1 (each 16-row tile is exclusive to one wave; reads precede writes).
__global__ __launch_bounds__(256) void k_gemm_fused(
    const float* A1, const float* A2,
    const float* __restrict__ W1, const float* __restrict__ W2,
    const float* __restrict__ bias, float* Out, int Mtiles) {
    // Swizzled B fragments: [mat(2)][ktile(32)][ntile(8)][lane(32)][pair(2)] f32 = 128KB
    __shared__ float sw[2 * 32 * 8 * 64];
    int tid = threadIdx.x;
    for (int f = tid; f < 32768; f += 256) {
        int p    = f & 1;
        int lane = (f >> 1) & 31;
        int nt   = (f >> 6) & 7;
        int kt   = (f >> 9) & 31;
        int m    = f >> 14;
        const float* W = m ? W2 : W1;
        int k = kt * 4 + ((lane >> 4) << 1) + p;   // lanes 16-31 hold K+2,K+3
        int c = nt * 16 + (lane & 15);
        sw[f] = W[k * FF + c];
    }
    __syncthreads();

    int lane = tid & 31;
    int mrow = lane & 15;
    int koff = (lane >> 4) << 1;
    int wid  = blockIdx.x * (blockDim.x >> 5) + (tid >> 5);
    int nw   = gridDim.x * (blockDim.x >> 5);
    const float* swA = sw;
    const float* swB = sw + 16384;

    for (int t = wid; t < Mtiles; t += nw) {
        long r0 = (long)t * 16 + mrow;
        const float* a1p = A1 + r0 * FF;
        const float* a2p = A2 + r0 * FF;
        v8f acc[8];
#pragma unroll
        for (int n = 0; n < 8; n++) {
            float bv = bias[n * 16 + mrow];
            acc[n] = (v8f){bv, bv, bv, bv, bv, bv, bv, bv};
        }
        for (int kt = 0; kt < 32; ++kt) {
            v2f a1 = *(const v2f*)(a1p + kt * 4 + koff);
            v2f a2 = *(const v2f*)(a2p + kt * 4 + koff);
            const float* b1b = swA + kt * 512 + lane * 2;
            const float* b2b = swB + kt * 512 + lane * 2;
#pragma unroll
            for (int n = 0; n < 8; n++) {
                v2f b1 = *(const v2f*)(b1b + n * 64);
                acc[n] = __builtin_amdgcn_wmma_f32_16x16x4_f32(
                    false, a1, false, b1, (short)0, acc[n], false, false);
            }
#pragma unroll
            for (int n = 0; n < 8; n++) {
                v2f b2 = *(const v2f*)(b2b + n * 64);
                acc[n] = __builtin_amdgcn_wmma_f32_16x16x4_f32(
                    false, a2, false, b2, (short)0, acc[n], false, false);
            }
        }
        long orow0 = (long)t * 16 + ((lane >> 4) << 3);  // C: hi lanes hold rows +8
#pragma unroll
        for (int n = 0; n < 8; n++) {
#pragma unroll
            for (int c = 0; c < 8; c++) {
                float v = acc[n][c];
                v = v > 0.f ? v : 0.f;
                Out[(orow0 + c) * FF + n * 16 + mrow] = v;
            }
        }
    }
}

// ------------------------------------------------------ pooling score
__global__ void k_score(const float* __restrict__ H, const float* __restrict__ pw,
                        float* __restrict__ score) {
    __shared__ float spw[FF];
    __shared__ float sinv;
    int tid = threadIdx.x;
    if (tid < FF) spw[tid] = pw[tid];
    __syncthreads();
    if (tid == 0) {
        float ss = 0.f;
        for (int k = 0; k < FF; k++) ss += spw[k] * spw[k];
        sinv = 1.0f / sqrtf(ss);
    }
    __syncthreads();
    int lane = tid & 31;
    int wid  = blockIdx.x * (blockDim.x >> 5) + (tid >> 5);
    int nw   = gridDim.x * (blockDim.x >> 5);
    float4 wv = ((const float4*)spw)[lane];
    for (int i = wid; i < NTN; i += nw) {
        float4 hv = ((const float4*)(H + (long)i * FF))[lane];
        float d = hv.x * wv.x + hv.y * wv.y + hv.z * wv.z + hv.w * wv.w;
        for (int off = 16; off; off >>= 1) d += __shfl_xor(d, off, 32);
        if (lane == 0) score[i] = tanhf(d * sinv);
    }
}

// --------------------------------------------- TopK set selection (per graph)
// rank by (score desc, index asc) == jax.lax.top_k tie-break; kept-set order is
// irrelevant downstream (aggregation + mean are permutation invariant).
__global__ __launch_bounds__(1024) void k_topk(const float* __restrict__ score,
                                               int* __restrict__ new_idx,
                                               int* __restrict__ perm,
                                               float* __restrict__ gsc) {
    __shared__ float ss[NN];
    __shared__ int cnt;
    int g = blockIdx.x, t = threadIdx.x;
    int node = g * NN + t;
    float s = score[node];
    ss[t] = s;
    if (t == 0) cnt = 0;
    __syncthreads();
    int rank = 0;
    for (int j = 0; j < NN; j++) {
        float sj = ss[j];
        rank += (sj > s) || (sj == s && j < t);
    }
    if (rank < KK) {
        int slot = atomicAdd(&cnt, 1);
        int ni = g * KK + slot;
        new_idx[node] = ni;
        perm[ni] = node;
        gsc[ni] = s;
    } else {
        new_idx[node] = -1;
    }
}

// ------------------------------------------------------ gated gather of kept rows
__global__ void k_gather_gate(const float* __restrict__ H, const int* __restrict__ perm,
                              const float* __restrict__ gsc, float* __restrict__ HP) {
    int lane = threadIdx.x & 31;
    int wid  = (blockIdx.x * blockDim.x + threadIdx.x) >> 5;
    int nw   = (gridDim.x * blockDim.x) >> 5;
    for (int r = wid; r < BKN; r += nw) {
        int node = perm[r];
        float s = gsc[r];
        float4 v = ((const float4*)(H + (long)node * FF))[lane];
        v.x *= s; v.y *= s; v.z *= s; v.w *= s;
        ((float4*)(HP + (long)r * FF))[lane] = v;
    }
}

// ---------------------------- head: mean-pool + MLP + log_softmax (per graph)
__global__ __launch_bounds__(128) void k_head(const float* __restrict__ H3,
                                              const float* __restrict__ W1,
                                              const float* __restrict__ b1,
                                              const float* __restrict__ W2,
                                              const float* __restrict__ b2,
                                              float* __restrict__ out) {
    __shared__ float gm[FF];
    __shared__ float g1[64];
    __shared__ float z[NCLS];
    __shared__ float red[2];
    int g = blockIdx.x, t = threadIdx.x;
    const float* base = H3 + (long)g * KK * FF;
    float acc = 0.f;
    for (int j = 0; j < KK; j++) acc += base[(long)j * FF + t];
    gm[t] = acc * (1.0f / KK);
    __syncthreads();
    if (t < 64) {
        float a = b1[t];
        for (int k = 0; k < FF; k++) a += gm[k] * W1[k * 64 + t];
        g1[t] = a > 0.f ? a : 0.f;
    }
    __syncthreads();
    if (t < NCLS) {
        float a = b2[t];
        for (int k = 0; k < 64; k++) a += g1[k] * W2[k * NCLS + t];
        z[t] = a;
    }
    __syncthreads();
    if (t == 0) {
        float m = z[0];
        for (int c = 1; c < NCLS; c++) m = fmaxf(m, z[c]);
        float se = 0.f;
        for (int c = 0; c < NCLS; c++) se += expf(z[c] - m);
        red[0] = m; red[1] = logf(se);
    }
    __syncthreads();
    if (t < NCLS) out[g * NCLS + t] = z[t] - red[0] - red[1];
}

// ---------------------------------------------------------------- launcher
extern "C" void kernel_launch(void* const* d_in, const int* in_sizes, int n_in,
                              void* d_out, int out_size, void* d_ws, size_t ws_size,
                              hipStream_t stream) {
    (void)in_sizes; (void)n_in; (void)out_size;
    const float* x   = (const float*)d_in[0];
    const int*   ei  = (const int*)d_in[1];
    const int*   src = ei;
    const int*   dst = ei + NE;
    const float* W1r = (const float*)d_in[3];
    const float* b1  = (const float*)d_in[4];
    const float* W1s = (const float*)d_in[5];
    const float* W2r = (const float*)d_in[6];
    const float* b2  = (const float*)d_in[7];
    const float* W2s = (const float*)d_in[8];
    const float* pw  = (const float*)d_in[9];
    const float* W3r = (const float*)d_in[10];
    const float* b3  = (const float*)d_in[11];
    const float* W3s = (const float*)d_in[12];
    const float* l1w = (const float*)d_in[13];
    const float* l1b = (const float*)d_in[14];
    const float* l2w = (const float*)d_in[15];
    const float* l2b = (const float*)d_in[16];

    const size_t SZ_BUF   = (size_t)NTN * FF * 4;   // 64 MB
    const size_t OFF_A    = 0;
    const size_t OFF_B    = SZ_BUF;
    const size_t OFF_SC   = 2 * SZ_BUF;
    const size_t OFF_GSC  = OFF_SC  + (size_t)NTN * 4;
    const size_t OFF_PERM = OFF_GSC + (size_t)BKN * 4;
    const size_t OFF_NIDX = OFF_PERM + (size_t)BKN * 4;
    const size_t TOTAL    = OFF_NIDX + (size_t)NTN * 4;
    if (ws_size < TOTAL) return;

    char* ws = (char*)d_ws;
    float* bufA  = (float*)(ws + OFF_A);    // agg1 -> h1 (in place) -> hp
    float* bufB  = (float*)(ws + OFF_B);    // agg2 -> h2 (in place) -> agg3 -> h3
    float* score = (float*)(ws + OFF_SC);
    float* gsc   = (float*)(ws + OFF_GSC);
    int*   perm  = (int*)(ws + OFF_PERM);
    int*   nidx  = (int*)(ws + OFF_NIDX);
    float* out   = (float*)d_out;

    // ---- layer 1: agg1 = scatter(x); h1 = relu(agg1@W1r + x@W1s + b1)
    k_zero<<<4096, 256, 0, stream>>>(bufA, (long)NTN * FF);
    k_scatter<<<8192, 256, 0, stream>>>(x, src, dst, bufA, NE);
    k_gemm_fused<<<1024, 256, 0, stream>>>(bufA, x, W1r, W1s, b1, bufA, NTN / 16);

    // ---- layer 2
    k_zero<<<4096, 256, 0, stream>>>(bufB, (long)NTN * FF);
    k_scatter<<<8192, 256, 0, stream>>>(bufA, src, dst, bufB, NE);
    k_gemm_fused<<<1024, 256, 0, stream>>>(bufB, bufA, W2r, W2s, b2, bufB, NTN / 16);

    // ---- TopK pooling
    k_score<<<2048, 256, 0, stream>>>(bufB, pw, score);
    k_topk<<<BG, 1024, 0, stream>>>(score, nidx, perm, gsc);
    k_gather_gate<<<4096, 256, 0, stream>>>(bufB, perm, gsc, bufA);  // hp -> bufA

    // ---- layer 3 on kept nodes
    k_zero<<<2048, 256, 0, stream>>>(bufB, (long)BKN * FF);
    k_scatter_masked<<<8192, 256, 0, stream>>>(bufA, src, dst, nidx, bufB, NE);
    k_gemm_fused<<<820, 256, 0, stream>>>(bufB, bufA, W3r, W3s, b3, bufB, BKN / 16);

    // ---- head
    k_head<<<BG, 128, 0, stream>>>(bufB, l1w, l1b, l2w, l2b, out);
}